// GNNModel_53549652246670
// MI455X (gfx1250) — compile-verified
//
#include <hip/hip_runtime.h>
#include <math.h>

#define N_NODES  50000
#define N_EDGES  800000
#define N_FEAT   128
#define HIDDEN   64
#define N_GRAPHS 256
#define M_TILES  (N_NODES / 16)   // 3125, exact

typedef float v2f __attribute__((ext_vector_type(2)));
typedef float v8f __attribute__((ext_vector_type(8)));

// ---------------------------------------------------------------------------
// Degree / norm kernels
// ---------------------------------------------------------------------------
__global__ void deg_init_kernel(float* __restrict__ deg) {
    int i = blockIdx.x * blockDim.x + threadIdx.x;
    if (i < N_NODES) deg[i] = 1.0f;   // self-loop contributes 1 to every node
}

__global__ void deg_accum_kernel(const int* __restrict__ ei, float* __restrict__ deg) {
    int e = blockIdx.x * blockDim.x + threadIdx.x;
    if (e < N_EDGES) atomicAdd(&deg[ei[N_EDGES + e]], 1.0f);   // dst row
}

__global__ void deg_rsqrt_kernel(float* __restrict__ deg) {
    int i = blockIdx.x * blockDim.x + threadIdx.x;
    if (i < N_NODES) {
        float d = deg[i];
        deg[i] = (d > 0.0f) ? rsqrtf(d) : 0.0f;
    }
}

// ---------------------------------------------------------------------------
// fp32 WMMA GEMM: Y[M x 64] = (RELU_IN ? relu(X) : X)[M x K] @ W[K x 64]
// One wave -> one 16-row tile, all 4 column tiles. 4 waves / block.
// V_WMMA_F32_16X16X4_F32: A,B = v2f per lane, C/D = v8f per lane.
// ---------------------------------------------------------------------------
template <int K, bool RELU_IN>
__global__ __launch_bounds__(128) void gemm_wmma_kernel(const float* __restrict__ X,
                                                        const float* __restrict__ W,
                                                        float* __restrict__ Y) {
    const int wave = threadIdx.x >> 5;
    const int lane = threadIdx.x & 31;
    const int tile = blockIdx.x * 4 + wave;
    if (tile >= M_TILES) return;

    const int m0    = tile * 16;
    const int khalf = (lane >> 4) << 1;        // 0 for lanes 0-15, 2 for lanes 16-31
    const int col   = lane & 15;

    const float* xrow = X + (size_t)(m0 + (lane & 15)) * K + khalf;

    v8f acc0 = {}, acc1 = {}, acc2 = {}, acc3 = {};

    for (int k0 = 0; k0 < K; k0 += 4) {
        v2f a;
        float a0 = xrow[k0];
        float a1 = xrow[k0 + 1];
        if (RELU_IN) { a0 = fmaxf(a0, 0.0f); a1 = fmaxf(a1, 0.0f); }
        a.x = a0; a.y = a1;

        const float* wk = W + (size_t)(k0 + khalf) * HIDDEN + col;
        v2f b0, b1, b2, b3;
        b0.x = wk[0];   b0.y = wk[HIDDEN + 0];
        b1.x = wk[16];  b1.y = wk[HIDDEN + 16];
        b2.x = wk[32];  b2.y = wk[HIDDEN + 32];
        b3.x = wk[48];  b3.y = wk[HIDDEN + 48];

        acc0 = __builtin_amdgcn_wmma_f32_16x16x4_f32(false, a, false, b0, (short)0, acc0, false, false);
        acc1 = __builtin_amdgcn_wmma_f32_16x16x4_f32(false, a, false, b1, (short)0, acc1, false, false);
        acc2 = __builtin_amdgcn_wmma_f32_16x16x4_f32(false, a, false, b2, (short)0, acc2, false, false);
        acc3 = __builtin_amdgcn_wmma_f32_16x16x4_f32(false, a, false, b3, (short)0, acc3, false, false);
    }

    // D layout: VGPR r -> row m0 + r (+8 for lanes 16-31), col = lane & 15
    const int orow = m0 + ((lane >> 4) << 3);
    float* y = Y + (size_t)orow * HIDDEN + col;
#pragma unroll
    for (int r = 0; r < 8; ++r) {
        y[(size_t)r * HIDDEN + 0]  = acc0[r];
        y[(size_t)r * HIDDEN + 16] = acc1[r];
        y[(size_t)r * HIDDEN + 32] = acc2[r];
        y[(size_t)r * HIDDEN + 48] = acc3[r];
    }
}

// ---------------------------------------------------------------------------
// agg[i] = h[i] * dis[i]^2 + bias   (self-loop message + bias, pre-scatter)
// 16 threads per node, float4 per thread.
// ---------------------------------------------------------------------------
__global__ void agg_init_kernel(const float* __restrict__ h,
                                const float* __restrict__ dis,
                                const float* __restrict__ bias,
                                float* __restrict__ agg) {
    int t  = blockIdx.x * blockDim.x + threadIdx.x;
    int i  = t >> 4;
    int f4 = (t & 15) << 2;
    if (i >= N_NODES) return;
    float d = dis[i];
    float s = d * d;
    float4 hv = *(const float4*)(h + (size_t)i * HIDDEN + f4);
    float4 o;
    o.x = hv.x * s + bias[f4 + 0];
    o.y = hv.y * s + bias[f4 + 1];
    o.z = hv.z * s + bias[f4 + 2];
    o.w = hv.w * s + bias[f4 + 3];
    *(float4*)(agg + (size_t)i * HIDDEN + f4) = o;
}

// ---------------------------------------------------------------------------
// Edge scatter: agg[dst] += h[src] * dis[src]*dis[dst]
// 16 threads per edge, float4 gather + 4 f32 atomics each.
// h (12.8 MB) lives in L2 -> gathers and atomics stay on-chip.
// ---------------------------------------------------------------------------
__global__ void edge_scatter_kernel(const int* __restrict__ ei,
                                    const float* __restrict__ h,
                                    const float* __restrict__ dis,
                                    float* __restrict__ agg) {
    int t  = blockIdx.x * blockDim.x + threadIdx.x;
    int e  = t >> 4;
    int f4 = (t & 15) << 2;
    if (e >= N_EDGES) return;
    int s = ei[e];
    int d = ei[N_EDGES + e];
    float nrm = dis[s] * dis[d];
    float4 hv = *(const float4*)(h + (size_t)s * HIDDEN + f4);
    float* ap = agg + (size_t)d * HIDDEN + f4;
    atomicAdd(ap + 0, hv.x * nrm);
    atomicAdd(ap + 1, hv.y * nrm);
    atomicAdd(ap + 2, hv.z * nrm);
    atomicAdd(ap + 3, hv.w * nrm);
}

// ---------------------------------------------------------------------------
// Pooling
// ---------------------------------------------------------------------------
__global__ void pool_zero_kernel(float* __restrict__ sums, float* __restrict__ cnts) {
    int i = blockIdx.x * blockDim.x + threadIdx.x;
    if (i < N_GRAPHS * HIDDEN) sums[i] = 0.0f;
    if (i < N_GRAPHS)          cnts[i] = 0.0f;
}

// sums[batch[i]] += relu(h[i]); cnts[batch[i]] += 1   (relu fused on load)
__global__ void pool_accum_kernel(const float* __restrict__ h,
                                  const int* __restrict__ batch,
                                  float* __restrict__ sums,
                                  float* __restrict__ cnts) {
    int t  = blockIdx.x * blockDim.x + threadIdx.x;
    int i  = t >> 4;
    int f4 = (t & 15) << 2;
    if (i >= N_NODES) return;
    int g = batch[i];
    float4 hv = *(const float4*)(h + (size_t)i * HIDDEN + f4);
    float* sp = sums + (size_t)g * HIDDEN + f4;
    atomicAdd(sp + 0, fmaxf(hv.x, 0.0f));
    atomicAdd(sp + 1, fmaxf(hv.y, 0.0f));
    atomicAdd(sp + 2, fmaxf(hv.z, 0.0f));
    atomicAdd(sp + 3, fmaxf(hv.w, 0.0f));
    if ((t & 15) == 0) atomicAdd(&cnts[g], 1.0f);
}

// ---------------------------------------------------------------------------
// MLP head: out[g] = relu(pooled[g] @ Wm1 + bm1) @ Wm2 + bm2
// One block of 256 threads; one graph per thread (64x64 FMAs each).
// ---------------------------------------------------------------------------
__global__ __launch_bounds__(256) void mlp_head_kernel(const float* __restrict__ sums,
                                                       const float* __restrict__ cnts,
                                                       const float* __restrict__ Wm1,
                                                       const float* __restrict__ bm1,
                                                       const float* __restrict__ Wm2,
                                                       const float* __restrict__ bm2,
                                                       float* __restrict__ out) {
    int g = threadIdx.x;
    if (g >= N_GRAPHS) return;
    float inv = 1.0f / fmaxf(cnts[g], 1.0f);
    float p[HIDDEN];
#pragma unroll
    for (int k = 0; k < HIDDEN; ++k) p[k] = sums[(size_t)g * HIDDEN + k] * inv;
    float o = bm2[0];
    for (int j = 0; j < HIDDEN; ++j) {
        float hj = bm1[j];
#pragma unroll
        for (int k = 0; k < HIDDEN; ++k) hj = fmaf(p[k], Wm1[(size_t)k * HIDDEN + j], hj);
        o = fmaf(fmaxf(hj, 0.0f), Wm2[j], o);
    }
    out[g] = o;
}

// ---------------------------------------------------------------------------
// Launch
// ---------------------------------------------------------------------------
extern "C" void kernel_launch(void* const* d_in, const int* in_sizes, int n_in,
                              void* d_out, int out_size, void* d_ws, size_t ws_size,
                              hipStream_t stream) {
    const float* x     = (const float*)d_in[0];
    const int*   ei    = (const int*)  d_in[1];
    const int*   batch = (const int*)  d_in[2];
    const float* W1    = (const float*)d_in[3];
    const float* b1    = (const float*)d_in[4];
    const float* W2    = (const float*)d_in[5];
    const float* b2    = (const float*)d_in[6];
    const float* Wm1   = (const float*)d_in[7];
    const float* bm1   = (const float*)d_in[8];
    const float* Wm2   = (const float*)d_in[9];
    const float* bm2   = (const float*)d_in[10];
    float* out = (float*)d_out;

    // Workspace layout (floats)
    float* ws   = (float*)d_ws;
    float* dis  = ws;                                   // N_NODES (deg -> dis in place)
    float* bufA = ws + 50048;                           // N_NODES*HIDDEN
    float* bufB = bufA + (size_t)N_NODES * HIDDEN;      // N_NODES*HIDDEN
    float* sums = bufB + (size_t)N_NODES * HIDDEN;      // N_GRAPHS*HIDDEN
    float* cnts = sums + (size_t)N_GRAPHS * HIDDEN;     // N_GRAPHS

    const int nodeBlk  = (N_NODES + 255) / 256;              // 196
    const int edgeBlk  = (N_EDGES + 255) / 256;              // 3125
    const int nodeFBlk = (N_NODES * 16 + 255) / 256;         // 3125
    const int edgeFBlk = (N_EDGES * 16 + 255) / 256;         // 50000
    const int gemmBlk  = (M_TILES + 3) / 4;                  // 782

    // deg -> dis
    deg_init_kernel <<<nodeBlk, 256, 0, stream>>>(dis);
    deg_accum_kernel<<<edgeBlk, 256, 0, stream>>>(ei, dis);
    deg_rsqrt_kernel<<<nodeBlk, 256, 0, stream>>>(dis);

    // Layer 1: h1 = x @ W1 ; agg1 = scatter(h1) + b1
    gemm_wmma_kernel<N_FEAT, false><<<gemmBlk, 128, 0, stream>>>(x, W1, bufA);
    agg_init_kernel    <<<nodeFBlk, 256, 0, stream>>>(bufA, dis, b1, bufB);
    edge_scatter_kernel<<<edgeFBlk, 256, 0, stream>>>(ei, bufA, dis, bufB);

    // Layer 2: h2 = relu(agg1) @ W2 (relu fused) ; agg2 = scatter(h2) + b2
    gemm_wmma_kernel<HIDDEN, true><<<gemmBlk, 128, 0, stream>>>(bufB, W2, bufA);
    agg_init_kernel    <<<nodeFBlk, 256, 0, stream>>>(bufA, dis, b2, bufB);
    edge_scatter_kernel<<<edgeFBlk, 256, 0, stream>>>(ei, bufA, dis, bufB);

    // Pool (relu fused on load) + MLP head
    pool_zero_kernel <<<(N_GRAPHS * HIDDEN + 255) / 256, 256, 0, stream>>>(sums, cnts);
    pool_accum_kernel<<<nodeFBlk, 256, 0, stream>>>(bufB, batch, sums, cnts);
    mlp_head_kernel  <<<1, 256, 0, stream>>>(sums, cnts, Wm1, bm1, Wm2, bm2, out);
}